// SparseGATLayer_3702261809872
// MI455X (gfx1250) — compile-verified
//
#include <hip/hip_runtime.h>
#include <hip/hip_bf16.h>
#include <math.h>

typedef __attribute__((ext_vector_type(2))) float v2f;
typedef __attribute__((ext_vector_type(8))) float v8f;
typedef __attribute__((ext_vector_type(4))) int   v4i;

#define LRELU_ALPHA 0.2f

// Monotone order-preserving float<->u32 mapping so segment-max can use
// the native global_atomic_max_u32 (deterministic, no CAS loop).
__device__ __forceinline__ unsigned encode_f32(float f) {
  unsigned u = __float_as_uint(f);
  return (u & 0x80000000u) ? ~u : (u | 0x80000000u);
}
__device__ __forceinline__ float decode_f32(unsigned u) {
  return (u & 0x80000000u) ? __uint_as_float(u & 0x7FFFFFFFu)
                           : __uint_as_float(~u);
}
#define NEG_INF_ENC 0x007FFFFFu  // encode_f32(-inf)

// ---------------------------------------------------------------------------
// Async (ASYNCcnt-tracked) global->LDS staging with compile-time fallback.
// Builtin signature (from hipcc diagnostic): (v4i AS1*, v4i AS3*, imm, imm).
// ---------------------------------------------------------------------------
#if defined(__gfx1250__) && __has_builtin(__builtin_amdgcn_global_load_async_to_lds_b128)
#define HAVE_ASYNC_LDS 1
typedef __attribute__((address_space(1))) v4i g1v4i;  // global b128 chunk
typedef __attribute__((address_space(3))) v4i l3v4i;  // LDS b128 chunk
#else
#define HAVE_ASYNC_LDS 0
#endif

__device__ __forceinline__ void wait_async_all() {
#if __has_builtin(__builtin_amdgcn_s_wait_asynccnt)
  __builtin_amdgcn_s_wait_asynccnt(0);
#else
  asm volatile("s_wait_asynccnt 0" ::: "memory");
#endif
}

// Copy one 32KB slab (2048 x b128) from global to LDS with 256 threads.
__device__ __forceinline__ void stage_slab(const v2f* __restrict__ gsrc,
                                           v2f* __restrict__ sdst) {
#if HAVE_ASYNC_LDS
  g1v4i* g = (g1v4i*)gsrc;
  l3v4i* l = (l3v4i*)sdst;
#pragma unroll
  for (int i = 0; i < 8; ++i) {
    const int chunk = threadIdx.x + i * 256;  // b128 units
    __builtin_amdgcn_global_load_async_to_lds_b128(g + chunk, l + chunk,
                                                   /*imm_offset=*/0, /*cpol=*/0);
  }
  wait_async_all();
#else
  float4* d4 = (float4*)sdst;
  const float4* s4 = (const float4*)gsrc;
#pragma unroll
  for (int i = 0; i < 8; ++i)
    d4[threadIdx.x + i * 256] = s4[threadIdx.x + i * 256];
#endif
}

// ---------------------------------------------------------------------------
// Kernel 0: swizzle W [256,64] row-major into K-pair-major fragments:
// Wsw[p][c] = (W[2p, c], W[2p+1, c]) so each WMMA B-fragment is one
// contiguous b64 in LDS (no repack movs in the GEMM inner loop).
// ---------------------------------------------------------------------------
__global__ void __launch_bounds__(256)
gat_wswizzle(const float* __restrict__ W, v2f* __restrict__ Wsw) {
  const int id = blockIdx.x * blockDim.x + threadIdx.x;  // 0..8191
  const int p = id >> 6;
  const int c = id & 63;
  v2f w;
  w.x = W[(2 * p) * 64 + c];
  w.y = W[(2 * p + 1) * 64 + c];
  Wsw[id] = w;
}

// ---------------------------------------------------------------------------
// Kernel 1: h = X @ W with V_WMMA_F32_16X16X4_F32.
// Block = 256 threads = 8 waves; each wave owns a 16-row M-tile and the full
// 64-wide N. Swizzled W staged through LDS in two 32KB slabs via async copy.
// ---------------------------------------------------------------------------
__global__ void __launch_bounds__(256)
gat_gemm_wmma(const float* __restrict__ X, const v2f* __restrict__ Wsw,
              float* __restrict__ h, int N) {
  __shared__ v2f sW[64 * 64];  // 32 KB slab: 64 K-pairs x 64 cols

  const int lane  = threadIdx.x & 31;
  const int wave  = threadIdx.x >> 5;
  const int mtile = blockIdx.x * 8 + wave;
  int mbase = mtile * 16;
  if (mbase > N - 16) mbase = N - 16;  // clamp: duplicate waves write same data

  const int mrow  = mbase + (lane & 15);
  const int khalf = lane >> 4;   // selects K pair {0,1} vs {2,3} within 4-step
  const int coln  = lane & 15;

  v8f acc0 = {}, acc1 = {}, acc2 = {}, acc3 = {};
  const float* __restrict__ Xrow = X + (size_t)mrow * 256;

  for (int kk = 0; kk < 2; ++kk) {
    __syncthreads();
    stage_slab(Wsw + (size_t)kk * 64 * 64, sW);
    __syncthreads();

#pragma unroll 4
    for (int kb = 0; kb < 32; ++kb) {
      const int lp = kb * 2 + khalf;                    // local K-pair index
      v2f a = *(const v2f*)(Xrow + kk * 128 + lp * 2);  // A frag: X[mrow, 2lp..2lp+1]
      const v2f* wrow = sW + lp * 64 + coln;
      v2f b0 = wrow[0];    // cols  0..15
      v2f b1 = wrow[16];   // cols 16..31
      v2f b2 = wrow[32];   // cols 32..47
      v2f b3 = wrow[48];   // cols 48..63
      acc0 = __builtin_amdgcn_wmma_f32_16x16x4_f32(false, a, false, b0, (short)0, acc0, false, false);
      acc1 = __builtin_amdgcn_wmma_f32_16x16x4_f32(false, a, false, b1, (short)0, acc1, false, false);
      acc2 = __builtin_amdgcn_wmma_f32_16x16x4_f32(false, a, false, b2, (short)0, acc2, false, false);
      acc3 = __builtin_amdgcn_wmma_f32_16x16x4_f32(false, a, false, b3, (short)0, acc3, false, false);
    }
  }

  // C/D layout: M = v + 8*(lane/16), N = lane%16 (+16/32/48 per tile)
  float* __restrict__ hp = h + (size_t)mbase * 64;
  const int mlo = (lane >> 4) * 8;
#pragma unroll
  for (int v = 0; v < 8; ++v) {
    const int m = mlo + v;
    hp[m * 64 + coln     ] = acc0[v];
    hp[m * 64 + coln + 16] = acc1[v];
    hp[m * 64 + coln + 32] = acc2[v];
    hp[m * 64 + coln + 48] = acc3[v];
  }
}

// ---------------------------------------------------------------------------
// Kernel 2: per-node alpha_src/alpha_dst (h . a halves), init rowmax/rowsum,
// zero d_out. One wave32 per node, each lane covers 2 of the 64 dims.
// ---------------------------------------------------------------------------
__global__ void __launch_bounds__(256)
gat_alpha_init(const float* __restrict__ h, const float* __restrict__ avec,
               float* __restrict__ alpha_s, float* __restrict__ alpha_d,
               unsigned* __restrict__ rowmax_enc, float* __restrict__ rowsum,
               float* __restrict__ out, int N) {
  const int row  = (blockIdx.x * blockDim.x + threadIdx.x) >> 5;
  const int lane = threadIdx.x & 31;
  if (row >= N) return;

  const float* hr = h + (size_t)row * 64;
  v2f hv = *(const v2f*)(hr + lane * 2);
  v2f a1 = *(const v2f*)(avec + lane * 2);
  v2f a2 = *(const v2f*)(avec + 64 + lane * 2);
  float s = hv.x * a1.x + hv.y * a1.y;
  float d = hv.x * a2.x + hv.y * a2.y;
#pragma unroll
  for (int off = 16; off > 0; off >>= 1) {
    s += __shfl_down(s, off, 32);
    d += __shfl_down(d, off, 32);
  }
  if (lane == 0) {
    alpha_s[row]    = s;
    alpha_d[row]    = d;
    rowmax_enc[row] = NEG_INF_ENC;
    rowsum[row]     = 0.0f;
  }
  // zero output row (harness poisons d_out; aggregation accumulates)
  *(v2f*)(out + (size_t)row * 64 + lane * 2) = (v2f){0.0f, 0.0f};
}

// ---------------------------------------------------------------------------
// Kernel 3: per-edge logit e = leakyrelu(alpha_s[src] + alpha_d[dst]);
// segment-max over src via encoded u32 atomicMax (global_atomic_max_u32).
// ---------------------------------------------------------------------------
__global__ void __launch_bounds__(256)
gat_edge_logit_max(const int* __restrict__ src, const int* __restrict__ dst,
                   const float* __restrict__ alpha_s, const float* __restrict__ alpha_d,
                   float* __restrict__ ebuf, unsigned* __restrict__ rowmax_enc, int E) {
  const int i = blockIdx.x * blockDim.x + threadIdx.x;
  if (i >= E) return;
  const int s = src[i];
  const int d = dst[i];
  float e = alpha_s[s] + alpha_d[d];
  e = (e > 0.0f) ? e : LRELU_ALPHA * e;
  ebuf[i] = e;
  atomicMax(rowmax_enc + s, encode_f32(e));
}

// ---------------------------------------------------------------------------
// Kernel 4: ex = exp(e - rowmax[src]); segment-sum via native f32 atomicAdd.
// ---------------------------------------------------------------------------
__global__ void __launch_bounds__(256)
gat_edge_expsum(const int* __restrict__ src, float* __restrict__ ebuf,
                const unsigned* __restrict__ rowmax_enc,
                float* __restrict__ rowsum, int E) {
  const int i = blockIdx.x * blockDim.x + threadIdx.x;
  if (i >= E) return;
  const int s = src[i];
  const float m  = decode_f32(rowmax_enc[s]);
  const float ex = __expf(ebuf[i] - m);
  ebuf[i] = ex;  // in-place: same element, no cross-thread hazard
  unsafeAtomicAdd(rowsum + s, ex);
}

// ---------------------------------------------------------------------------
// Kernel 5: out[src] += (ex/rowsum[src]) * h[dst]. One wave32 per edge,
// 2 dims per lane -> global_atomic_add_f32 into L2-resident rows.
// ---------------------------------------------------------------------------
__global__ void __launch_bounds__(256)
gat_aggregate(const int* __restrict__ src, const int* __restrict__ dst,
              const float* __restrict__ ebuf, const float* __restrict__ rowsum,
              const float* __restrict__ h, float* __restrict__ out, int E) {
  const int e    = (blockIdx.x * blockDim.x + threadIdx.x) >> 5;
  const int lane = threadIdx.x & 31;
  if (e >= E) return;
  const int s = src[e];
  const int d = dst[e];
  const float attn = ebuf[e] / rowsum[s];
  v2f hv = *(const v2f*)(h + (size_t)d * 64 + lane * 2);
  float* os = out + (size_t)s * 64 + lane * 2;
  unsafeAtomicAdd(os,     attn * hv.x);
  unsafeAtomicAdd(os + 1, attn * hv.y);
}

// ---------------------------------------------------------------------------
static inline size_t align256(size_t x) { return (x + 255) & ~(size_t)255; }

extern "C" void kernel_launch(void* const* d_in, const int* in_sizes, int n_in,
                              void* d_out, int out_size, void* d_ws, size_t ws_size,
                              hipStream_t stream) {
  const float* X    = (const float*)d_in[0];
  const int*   edges= (const int*)  d_in[1];
  const float* W    = (const float*)d_in[2];
  const float* avec = (const float*)d_in[3];

  const int N = in_sizes[0] / 256;
  const int E = in_sizes[1] / 2;
  const int* src = edges;       // edges[0, :]
  const int* dst = edges + E;   // edges[1, :]
  float* out = (float*)d_out;

  char* ws = (char*)d_ws;
  size_t off = 0;
  float*    h       = (float*)(ws + off);    off += align256((size_t)N * 64 * 4);
  float*    ebuf    = (float*)(ws + off);    off += align256((size_t)E * 4);
  float*    alpha_s = (float*)(ws + off);    off += align256((size_t)N * 4);
  float*    alpha_d = (float*)(ws + off);    off += align256((size_t)N * 4);
  unsigned* rmax    = (unsigned*)(ws + off); off += align256((size_t)N * 4);
  float*    rsum    = (float*)(ws + off);    off += align256((size_t)N * 4);
  v2f*      Wsw     = (v2f*)(ws + off);      off += align256((size_t)128 * 64 * 8);

  // 0) swizzle W into K-pair-major fragment layout (8192 pairs)
  gat_wswizzle<<<32, 256, 0, stream>>>(W, Wsw);

  // 1) h = X @ W  (WMMA f32, async LDS staging of Wsw)
  const int mtiles = (N + 15) / 16;
  gat_gemm_wmma<<<(mtiles + 7) / 8, 256, 0, stream>>>(X, Wsw, h, N);

  // 2) per-node alphas + init rowmax/rowsum + zero out
  gat_alpha_init<<<((size_t)N * 32 + 255) / 256, 256, 0, stream>>>(
      h, avec, alpha_s, alpha_d, rmax, rsum, out, N);

  // 3) edge logits + segment max
  gat_edge_logit_max<<<(E + 255) / 256, 256, 0, stream>>>(
      src, dst, alpha_s, alpha_d, ebuf, rmax, E);

  // 4) exp + segment sum
  gat_edge_expsum<<<(E + 255) / 256, 256, 0, stream>>>(src, ebuf, rmax, rsum, E);

  // 5) weighted aggregation
  gat_aggregate<<<((size_t)E * 32 + 255) / 256, 256, 0, stream>>>(
      src, dst, ebuf, rsum, h, out, E);
}